// SelfContact_86947317940880
// MI455X (gfx1250) — compile-verified
//
#include <hip/hip_runtime.h>
#include <hip/hip_bf16.h>

#define NV 6890
#define NF 13776
#define EUCL_THRES 0.02f
#define WN_THRES 0.99f

typedef __attribute__((ext_vector_type(2))) float v2f;
typedef __attribute__((ext_vector_type(4))) float v4f;
typedef __attribute__((ext_vector_type(8))) float v8f;

#define NJT   ((NV + 15) / 16)           // 431 column tiles of 16
#define NC32  ((NV + 31) / 32)           // 216 row chunks of 32
#define ISEG  8                          // row-segments per column strip
#define CPS   ((NC32 + ISEG - 1) / ISEG) // 27 chunks per segment
#define TRI_TILE 512
#define NTC   ((NF + TRI_TILE - 1) / TRI_TILE) // 27 triangle chunks

// ---------------------------------------------------------------------------
// Kernel 0: pack xq[i]=(x,y,z,|x|^2), gather padded triangles (12 floats/tri),
// init argmin keys, bit-pack geomask: gmp[w*NV+j] bit b = gm[w*32+b][j].
// ---------------------------------------------------------------------------
__global__ void prep_kernel(const float* __restrict__ x,
                            const int*   __restrict__ faces,
                            const unsigned char* __restrict__ gm,
                            float* __restrict__ xq,    // [NV*4]
                            float* __restrict__ tris,  // [NF*12]
                            unsigned* __restrict__ gmp,
                            unsigned long long* __restrict__ minkey)
{
    int tid = blockIdx.x * blockDim.x + threadIdx.x;
    if (tid < NF) {
        #pragma unroll
        for (int k = 0; k < 3; ++k) {
            int vi = faces[tid * 3 + k];
            tris[tid * 12 + k * 3 + 0] = x[vi * 3 + 0];
            tris[tid * 12 + k * 3 + 1] = x[vi * 3 + 1];
            tris[tid * 12 + k * 3 + 2] = x[vi * 3 + 2];
        }
    }
    if (tid < NV) {
        float a = x[tid * 3 + 0], b = x[tid * 3 + 1], c = x[tid * 3 + 2];
        xq[tid * 4 + 0] = a;
        xq[tid * 4 + 1] = b;
        xq[tid * 4 + 2] = c;
        xq[tid * 4 + 3] = a * a + b * b + c * c;
        minkey[tid] = ~0ULL;
    }
    if (tid < NC32 * NV) {
        int w = tid / NV;
        int j = tid - w * NV;
        unsigned m = 0;
        #pragma unroll 4
        for (int b = 0; b < 32; ++b) {
            int i  = w * 32 + b;
            int ic = (i < NV) ? i : (NV - 1);
            unsigned bit = (i < NV) && (gm[(size_t)ic * NV + j] != 0);
            m |= bit << b;
        }
        gmp[(size_t)w * NV + j] = m;
    }
}

// ---------------------------------------------------------------------------
// Kernel 1: winding-number partials. Block (bx, chunk): 256 points x 512 tris.
// Triangles padded to 12 floats -> aligned b128 LDS broadcasts.
// ---------------------------------------------------------------------------
__global__ void winding_partial(const float* __restrict__ x,
                                const float* __restrict__ tris, // [NF*12]
                                float* __restrict__ wn_part)    // [NTC*NV]
{
    __shared__ float lt[TRI_TILE * 12]; // 24 KB

    int tid   = blockIdx.x * blockDim.x + threadIdx.x;
    int chunk = blockIdx.y;
    int t0    = chunk * TRI_TILE;
    int cnt   = (NF - t0) < TRI_TILE ? (NF - t0) : TRI_TILE;

    int pidx = tid < NV ? tid : NV - 1;
    float px = x[pidx * 3 + 0];
    float py = x[pidx * 3 + 1];
    float pz = x[pidx * 3 + 2];

    for (int k = threadIdx.x; k < cnt * 12; k += blockDim.x)
        lt[k] = tris[(size_t)t0 * 12 + k];
    __syncthreads();

    float acc = 0.0f;
    for (int t = 0; t < cnt; ++t) {
        const float* tr = &lt[t * 12]; // uniform addr -> LDS broadcast
        float rax = tr[0] - px, ray = tr[1] - py, raz = tr[2] - pz;
        float rbx = tr[3] - px, rby = tr[4] - py, rbz = tr[5] - pz;
        float rcx = tr[6] - px, rcy = tr[7] - py, rcz = tr[8] - pz;

        // raw v_sqrt_f32 (~1 ulp): error << summation-order drift vs reference
        float la = __builtin_amdgcn_sqrtf(rax * rax + ray * ray + raz * raz);
        float lb = __builtin_amdgcn_sqrtf(rbx * rbx + rby * rby + rbz * rbz);
        float lc = __builtin_amdgcn_sqrtf(rcx * rcx + rcy * rcy + rcz * rcz);

        float cx = rby * rcz - rbz * rcy;
        float cy = rbz * rcx - rbx * rcz;
        float cz = rbx * rcy - rby * rcx;

        float det = rax * cx + ray * cy + raz * cz;
        float dab = rax * rbx + ray * rby + raz * rbz;
        float dbc = rbx * rcx + rby * rcy + rbz * rcz;
        float dca = rcx * rax + rcy * ray + rcz * raz;

        float denom = la * lb * lc + dab * lc + dbc * la + dca * lb;
        acc += atan2f(det, denom);
    }

    if (tid < NV)
        wn_part[(size_t)chunk * NV + tid] = acc;
}

// ---------------------------------------------------------------------------
// Kernel 2: distance argmin. A-row=(x,y,z,sq_i), B-col=(-2x_j,-2y_j,-2z_j,1)
// => WMMA C[i][j] = sq_i - 2*dot  (full K=4 used, nothing wasted).
// Inner loop per element: cmp + 2 cndmask. No LDS, no sqrt, no fma.
// ---------------------------------------------------------------------------
__global__ void dist_partial(const float* __restrict__ xq,   // [NV*4]
                             const unsigned* __restrict__ gmp,
                             unsigned long long* __restrict__ minkey)
{
    int wave = threadIdx.x >> 5;
    int lane = threadIdx.x & 31;
    int gw   = blockIdx.x * 8 + wave;
    int jtile = gw >> 3;                 // / ISEG
    int iseg  = gw & 7;                  // % ISEG
    if (jtile >= NJT) return;            // uniform per wave

    int  nn = lane & 15;
    bool hi = lane >= 16;
    int  j  = jtile * 16 + nn;
    int  jc = (j < NV) ? j : (NV - 1);

    // B fragment (4x16), fixed per strip. K = hi ? {2,3} : {0,1}
    v4f qj = *(const v4f*)(xq + (size_t)jc * 4);
    v2f bfrag;
    bfrag[0] = hi ? (-2.0f * qj[2]) : (-2.0f * qj[0]);
    bfrag[1] = hi ? 1.0f            : (-2.0f * qj[1]);
    float sqj = qj[3];

    float minc   = __builtin_inff();   // min of (sq_i - 2*dot); sq_j added later
    int   minidx = 0;

    int c0 = iseg * CPS;
    int c1 = (c0 + CPS < NC32) ? (c0 + CPS) : NC32;

    for (int c = c0; c < c1; ++c) {
        unsigned mword = gmp[(size_t)c * NV + jc];
        int cp = (c + 1 < NC32) ? (c + 1) : (NC32 - 1);
        __builtin_prefetch(gmp + (size_t)cp * NV + jc, 0, 3); // global_prefetch_b8

        #pragma unroll
        for (int half = 0; half < 2; ++half) {
            int ibase = (2 * c + half) * 16;
            int imc   = ibase + nn;
            imc = (imc < NV) ? imc : (NV - 1);   // clamped rows are masked out

            // A fragment (16x4): one aligned b128 per lane
            v4f qi = *(const v4f*)(xq + (size_t)imc * 4);
            v2f afrag;
            afrag[0] = hi ? qi[2] : qi[0];
            afrag[1] = hi ? qi[3] : qi[1];       // K=3 slot carries sq_i

            v8f cf = {};
            cf = __builtin_amdgcn_wmma_f32_16x16x4_f32(
                    false, afrag, false, bfrag, (short)0, cf, false, false);

            int rbase = ibase + (hi ? 8 : 0);
            int bbase = half * 16 + (hi ? 8 : 0);
            #pragma unroll
            for (int v = 0; v < 8; ++v) {
                int  i    = rbase + v;
                bool ok   = (mword >> (bbase + v)) & 1u; // 0 past NV (padded)
                bool take = ok && (cf[v] < minc);
                minidx = take ? i : minidx;
                minc   = take ? cf[v] : minc;
            }
        }
    }

    // Combine half-wave partials per column (wave32 shuffle).
    float oval = __shfl_xor(minc, 16, 32);
    int   oidx = __shfl_xor(minidx, 16, 32);
    bool  take = (oval < minc) || (oval == minc && oidx < minidx);
    minc   = take ? oval : minc;
    minidx = take ? oidx : minidx;

    if (!hi && j < NV) {
        float d2c = fmaxf(minc + sqj, 0.0f);     // >= 0 -> bits are order-monotone
        unsigned long long key =
            ((unsigned long long)__float_as_uint(d2c) << 32) | (unsigned)minidx;
        atomicMin(minkey + j, key);              // order-independent => deterministic
    }
}

// ---------------------------------------------------------------------------
// Kernel 3: finalize — fixed-order wn reduction + argmin unpack.
// ---------------------------------------------------------------------------
__global__ void finalize_kernel(const float* __restrict__ xq,
                                const float* __restrict__ wn_part,
                                const unsigned long long* __restrict__ minkey,
                                float* __restrict__ out)
{
    int j = blockIdx.x * blockDim.x + threadIdx.x;
    if (j >= NV) return;

    float acc = 0.0f;
    for (int c = 0; c < NTC; ++c)          // fixed order => deterministic
        acc += wn_part[(size_t)c * NV + j];
    float wn = acc * 0.15915494309189535f; // 1/(2*pi)
    out[2 * NV + j] = (wn <= WN_THRES) ? 1.0f : 0.0f;

    unsigned long long key = minkey[j];
    int idx = (int)(unsigned)key;
    if (idx >= NV || idx < 0) idx = 0;

    // v2v_min[j] = unmasked v2v[idx, j], recomputed at full precision
    v4f qi = *(const v4f*)(xq + (size_t)idx * 4);
    v4f qj = *(const v4f*)(xq + (size_t)j * 4);
    float dot = qi[0] * qj[0] + qi[1] * qj[1] + qi[2] * qj[2];
    float d2  = qi[3] + qj[3] - 2.0f * dot;
    float v   = sqrtf(fmaxf(d2, 0.0f));
    out[j]      = v;
    out[NV + j] = (v < EUCL_THRES) ? 1.0f : 0.0f;
}

// ---------------------------------------------------------------------------
extern "C" void kernel_launch(void* const* d_in, const int* in_sizes, int n_in,
                              void* d_out, int out_size, void* d_ws, size_t ws_size,
                              hipStream_t stream) {
    const float*         x     = (const float*)d_in[0];
    const int*           faces = (const int*)d_in[1];
    const unsigned char* gm    = (const unsigned char*)d_in[2];
    float* out = (float*)d_out;

    // Workspace layout (floats unless noted); 16B alignment for xq (b128 loads),
    // 8B alignment for minkey.
    float* xq      = (float*)d_ws;                          // NV*4   = 27560
    float* tris    = xq + (size_t)NV * 4;                   // NF*12  = 165312
    float* wn_part = tris + (size_t)NF * 12;                // NTC*NV = 186030
    unsigned long long* minkey =
        (unsigned long long*)(wn_part + (size_t)NTC * NV);  // NV u64 (offset 378902 floats, even)
    unsigned* gmp = (unsigned*)(minkey + NV);               // NC32*NV u32

    int prep_items  = NC32 * NV;
    int prep_blocks = (prep_items + 255) / 256;
    prep_kernel<<<prep_blocks, 256, 0, stream>>>(x, faces, gm, xq, tris, gmp, minkey);

    dim3 wgrid((NV + 255) / 256, NTC);
    winding_partial<<<wgrid, 256, 0, stream>>>(x, tris, wn_part);

    int dist_blocks = (NJT * ISEG + 7) / 8;                 // 8 waves per block
    dist_partial<<<dist_blocks, 256, 0, stream>>>(xq, gmp, minkey);

    finalize_kernel<<<(NV + 255) / 256, 256, 0, stream>>>(xq, wn_part, minkey, out);
}